// Autoencoder_60842506715667
// MI455X (gfx1250) — compile-verified
//
#include <hip/hip_runtime.h>
#include <math.h>

// Problem constants (from reference)
#define BB 8
#define VV 2048
#define NB 20
#define BV (BB*VV)

typedef __attribute__((ext_vector_type(2))) float v2f;
typedef __attribute__((ext_vector_type(8))) float v8f;

// ---------------------------------------------------------------------------
// CDNA5 WMMA f32 16x16x4 wrapper (D = A(16x4) * B(4x16) + C)
// A layout: lane l, m=l&15 : lanes 0-15 hold K=0 (x) / K=1 (y), lanes 16-31
// hold K=2 (z) / K=3 (pad 0).  B layout mirrors with n=l&15.
// C/D: VGPR r, lane l -> M = r + 8*(l>=16), N = l&15.
// ---------------------------------------------------------------------------
__device__ __forceinline__ v8f wmma_f32_16x16x4(v2f a, v2f b, v8f c) {
  return __builtin_amdgcn_wmma_f32_16x16x4_f32(
      /*neg_a=*/false, a, /*neg_b=*/false, b,
      /*c_mod=*/(short)0, c, /*reuse_a=*/false, /*reuse_b=*/false);
}

// ---------------------------------------------------------------------------
// Normalize columns of a 3 x cols matrix (axis-0 L2 norm, clamp 1e-12)
// ---------------------------------------------------------------------------
__global__ void normalize_cols_kernel(const float* __restrict__ src,
                                      float* __restrict__ dst, int cols) {
  int j = blockIdx.x * blockDim.x + threadIdx.x;
  if (j >= cols) return;
  float x = src[j], y = src[cols + j], z = src[2 * cols + j];
  float n = fmaxf(sqrtf(x * x + y * y + z * z), 1e-12f);
  dst[j] = x / n;
  dst[cols + j] = y / n;
  dst[2 * cols + j] = z / n;
}

// ---------------------------------------------------------------------------
// kNN: one block = one (batch, 16-row band).  8 waves split the 128 column
// tiles; each tile's 16x16 inner products come from one f32 WMMA.  The
// 16 x 2048 distance band lives in (dynamic) LDS, then each wave extracts
// the 20 nearest for 2 rows with an iterative wave32 argmin.
// ---------------------------------------------------------------------------
__global__ void knn_kernel(const float* __restrict__ verts,
                           int* __restrict__ idx_out) {
  extern __shared__ float smem[];          // 16*VV dist + 16 row-sq
  float* dist = smem;
  float* rowsq = smem + 16 * VV;

  const int b = blockIdx.x / (VV / 16);
  const int rt = blockIdx.x % (VV / 16);
  const int r0 = rt * 16;
  const int tid = threadIdx.x;
  const int lane = tid & 31;
  const int wave = tid >> 5;
  const bool hi = lane >= 16;
  const float* vb = verts + (size_t)b * VV * 3;

  if (tid < 16) {
    float x = vb[(r0 + tid) * 3 + 0];
    float y = vb[(r0 + tid) * 3 + 1];
    float z = vb[(r0 + tid) * 3 + 2];
    rowsq[tid] = x * x + y * y + z * z;
  }
  __syncthreads();

  // Build A operand (shared by every column tile of this wave)
  const int m = lane & 15;
  float ax = vb[(r0 + m) * 3 + 0];
  float ay = vb[(r0 + m) * 3 + 1];
  float az = vb[(r0 + m) * 3 + 2];
  v2f a;
  if (!hi) { a.x = ax; a.y = ay; } else { a.x = az; a.y = 0.0f; }

  float rs[8];
#pragma unroll
  for (int r = 0; r < 8; ++r) rs[r] = rowsq[r + (hi ? 8 : 0)];

  for (int ct = wave; ct < VV / 16; ct += 8) {
    const int c0 = ct * 16;
    const int n = lane & 15;
    float bx = vb[(c0 + n) * 3 + 0];
    float by = vb[(c0 + n) * 3 + 1];
    float bz = vb[(c0 + n) * 3 + 2];
    v2f bo;
    if (!hi) { bo.x = bx; bo.y = by; } else { bo.x = bz; bo.y = 0.0f; }
    const float csq = bx * bx + by * by + bz * bz;

    v8f c = {};
    c = wmma_f32_16x16x4(a, bo, c);

#pragma unroll
    for (int r = 0; r < 8; ++r) {
      const int row = r + (hi ? 8 : 0);
      float d = rs[r] + csq - 2.0f * c[r];
      if (r0 + row == c0 + n) d = 3.0e38f;   // exclude self
      dist[row * VV + c0 + n] = d;
    }
  }
  __syncthreads();

  // Top-NB per row: wave w owns rows 2w and 2w+1
  for (int rr = 0; rr < 2; ++rr) {
    const int row = wave * 2 + rr;
    float* drow = dist + row * VV;
    for (int k = 0; k < NB; ++k) {
      float best = 3.0e38f;
      int bi = VV;
      for (int j = lane; j < VV; j += 32) {
        float dv = drow[j];
        if (dv < best || (dv == best && j < bi)) { best = dv; bi = j; }
      }
#pragma unroll
      for (int off = 16; off > 0; off >>= 1) {
        float ob = __shfl_xor(best, off, 32);
        int oi = __shfl_xor(bi, off, 32);
        if (ob < best || (ob == best && oi < bi)) { best = ob; bi = oi; }
      }
      if (lane == 0) {
        idx_out[((size_t)b * VV + r0 + row) * NB + k] = bi;
        drow[bi] = 3.0e38f;                 // consume
      }
      asm volatile("s_wait_dscnt 0" ::: "memory");
    }
  }
}

// ---------------------------------------------------------------------------
// Specialized GEMM  Y[M x N] = X[M x K] @ W[K x N] + bias[N]
// One wave32 per 16x16 output tile via f32 WMMA.  K, N compile-time so the
// WMMA chain fully unrolls (guards for K%4 / N%16 fold to lane selects).
// M is always a multiple of 16 here.
// ---------------------------------------------------------------------------
template <int K, int N>
__global__ void __launch_bounds__(32)
gemm_bias_kernel(const float* __restrict__ X, const float* __restrict__ W,
                 const float* __restrict__ bias, float* __restrict__ Y) {
  constexpr int NT = (N + 15) / 16;
  const int mt = blockIdx.x / NT;
  const int nt = blockIdx.x % NT;
  const int lane = threadIdx.x;
  const bool hi = lane >= 16;
  const int m = mt * 16 + (lane & 15);
  const int n = nt * 16 + (lane & 15);
  const bool nok = (N % 16 == 0) ? true : (n < N);

  v8f acc = {};
#pragma unroll
  for (int k0 = 0; k0 < K; k0 += 4) {
    const int ka = k0 + (hi ? 2 : 0);
    v2f a, bo;
    a.x = (ka < K) ? X[(size_t)m * K + ka] : 0.0f;
    a.y = (ka + 1 < K) ? X[(size_t)m * K + ka + 1] : 0.0f;
    bo.x = (ka < K && nok) ? W[(size_t)ka * N + n] : 0.0f;
    bo.y = (ka + 1 < K && nok) ? W[(size_t)(ka + 1) * N + n] : 0.0f;
    acc = wmma_f32_16x16x4(a, bo, acc);
  }
  if (nok) {
    const float bv = bias[n];
#pragma unroll
    for (int r = 0; r < 8; ++r) {
      const int row = mt * 16 + r + (hi ? 8 : 0);
      Y[(size_t)row * N + n] = acc[r] + bv;
    }
  }
}

// ---------------------------------------------------------------------------
// conv_surface: f1[v][k] = relu( sum_s max_n relu(dn(v,n) . dir[:, s*16+k]) )
// block: 16 vertices x 16 channels; dn staged in LDS
// ---------------------------------------------------------------------------
__global__ void __launch_bounds__(256)
conv_surface_kernel(const float* __restrict__ verts, const int* __restrict__ idx,
                    const float* __restrict__ dirsN, float* __restrict__ f1) {
  __shared__ float dn[16 * NB * 3];
  const int v0 = blockIdx.x * 16;
  const int tid = threadIdx.x;

  for (int e = tid; e < 16 * NB; e += 256) {
    const int lv = e / NB, n = e - lv * NB;
    const int vflat = v0 + lv;
    const int b = vflat / VV;
    const int vloc = vflat - b * VV;
    const float* vb = verts + (size_t)b * VV * 3;
    const int nb = idx[(size_t)vflat * NB + n];
    float dx = vb[nb * 3 + 0] - vb[vloc * 3 + 0];
    float dy = vb[nb * 3 + 1] - vb[vloc * 3 + 1];
    float dz = vb[nb * 3 + 2] - vb[vloc * 3 + 2];
    float nn = fmaxf(sqrtf(dx * dx + dy * dy + dz * dz), 1e-12f);
    dn[e * 3 + 0] = dx / nn;
    dn[e * 3 + 1] = dy / nn;
    dn[e * 3 + 2] = dz / nn;
  }
  __syncthreads();

  const int lv = tid >> 4, k = tid & 15;
  const float* dnv = dn + lv * NB * 3;
  float acc = 0.0f;
#pragma unroll
  for (int s = 0; s < 4; ++s) {
    const int j = s * 16 + k;
    const float d0 = dirsN[j], d1 = dirsN[64 + j], d2 = dirsN[128 + j];
    float mx = 0.0f;   // theta >= 0 after relu
    for (int n = 0; n < NB; ++n) {
      float th = fmaxf(d0 * dnv[n * 3] + d1 * dnv[n * 3 + 1] + d2 * dnv[n * 3 + 2], 0.0f);
      mx = fmaxf(mx, th);
    }
    acc += mx;
  }
  f1[(size_t)(v0 + lv) * 16 + k] = fmaxf(acc, 0.0f);
}

// ---------------------------------------------------------------------------
// conv_layer: out[v][k] = act( feat[v][k] +
//     sum_s max_n ( relu(dn . dirN[:, s*outc+k]) * feat[nbr][outc + s*outc+k] ) )
// blockDim = (outc, vpb); dn staged in dynamic LDS (vpb*NB*3 floats)
// act: 0=none 1=relu 2=sigmoid
// Prefetches the next neighbor's feature row (global_prefetch_b8) to hide the
// random-gather latency against the 192MB L2.
// ---------------------------------------------------------------------------
__global__ void conv_layer_kernel(const float* __restrict__ verts,
                                  const int* __restrict__ idx,
                                  const float* __restrict__ dirsN,
                                  const float* __restrict__ feat,
                                  float* __restrict__ out, int act) {
  extern __shared__ float sdn[];
  const int outc = blockDim.x;
  const int vpb = blockDim.y;
  const int v0 = blockIdx.x * vpb;
  const int tid = threadIdx.y * outc + threadIdx.x;
  const int nth = outc * vpb;

  for (int e = tid; e < vpb * NB; e += nth) {
    const int lv = e / NB, n = e - lv * NB;
    const int vflat = v0 + lv;
    const int b = vflat / VV;
    const int vloc = vflat - b * VV;
    const float* vb = verts + (size_t)b * VV * 3;
    const int nb = idx[(size_t)vflat * NB + n];
    float dx = vb[nb * 3 + 0] - vb[vloc * 3 + 0];
    float dy = vb[nb * 3 + 1] - vb[vloc * 3 + 1];
    float dz = vb[nb * 3 + 2] - vb[vloc * 3 + 2];
    float nn = fmaxf(sqrtf(dx * dx + dy * dy + dz * dz), 1e-12f);
    sdn[e * 3 + 0] = dx / nn;
    sdn[e * 3 + 1] = dy / nn;
    sdn[e * 3 + 2] = dz / nn;
  }
  __syncthreads();

  const int lv = threadIdx.y, k = threadIdx.x;
  const int vflat = v0 + lv;
  const int b = vflat / VV;
  const int Ctot = 5 * outc;         // (SUP+1)*outc
  const int K4 = 4 * outc;
  const float* dnv = sdn + lv * NB * 3;

  float acc = feat[(size_t)vflat * Ctot + k];   // center
  float mx[4] = {-3.0e38f, -3.0e38f, -3.0e38f, -3.0e38f};

  int nb_cur = idx[(size_t)vflat * NB + 0];
  for (int n = 0; n < NB; ++n) {
    // prefetch next neighbor's support row while processing this one
    if (n + 1 < NB) {
      const int nb_next = idx[(size_t)vflat * NB + n + 1];
      __builtin_prefetch(feat + (size_t)(b * VV + nb_next) * Ctot + outc + k, 0, 3);
    }
    const float dx = dnv[n * 3 + 0];
    const float dy = dnv[n * 3 + 1];
    const float dz = dnv[n * 3 + 2];
    const int nbv = b * VV + nb_cur;
    if (n + 1 < NB) nb_cur = idx[(size_t)vflat * NB + n + 1];
    const float* sf = feat + (size_t)nbv * Ctot + outc;
#pragma unroll
    for (int s = 0; s < 4; ++s) {
      const int j = s * outc + k;
      float th = fmaxf(dx * dirsN[j] + dy * dirsN[K4 + j] + dz * dirsN[2 * K4 + j], 0.0f);
      float p = th * sf[j];
      mx[s] = fmaxf(mx[s], p);
    }
  }
  acc += mx[0] + mx[1] + mx[2] + mx[3];
  if (act == 1) acc = fmaxf(acc, 0.0f);
  else if (act == 2) acc = 1.0f / (1.0f + expf(-acc));
  out[(size_t)vflat * outc + k] = acc;
}

// ---------------------------------------------------------------------------
// AdaIN statistics: one block per (b, ch); mean + ddof=1 std over V verts
// ---------------------------------------------------------------------------
__global__ void __launch_bounds__(256)
adain_stats_kernel(const float* __restrict__ x, float* __restrict__ stats) {
  const int b = blockIdx.x / 32;
  const int c = blockIdx.x % 32;
  __shared__ float s1s[256], s2s[256];
  float s1 = 0.0f, s2 = 0.0f;
  for (int v = threadIdx.x; v < VV; v += 256) {
    float val = x[((size_t)b * VV + v) * 32 + c];
    s1 += val;
    s2 += val * val;
  }
  s1s[threadIdx.x] = s1;
  s2s[threadIdx.x] = s2;
  __syncthreads();
  for (int off = 128; off > 0; off >>= 1) {
    if (threadIdx.x < off) {
      s1s[threadIdx.x] += s1s[threadIdx.x + off];
      s2s[threadIdx.x] += s2s[threadIdx.x + off];
    }
    __syncthreads();
  }
  if (threadIdx.x == 0) {
    float mean = s1s[0] / (float)VV;
    float var = (s2s[0] - (float)VV * mean * mean) / (float)(VV - 1);
    stats[(b * 32 + c) * 2 + 0] = mean;
    stats[(b * 32 + c) * 2 + 1] = sqrtf(fmaxf(var, 0.0f)) + 1e-8f;
  }
}

__global__ void adain_apply_kernel(const float* __restrict__ x,
                                   const float* __restrict__ h,
                                   const float* __restrict__ stats,
                                   float* __restrict__ t) {
  const int i = blockIdx.x * 256 + threadIdx.x;
  if (i >= BV * 32) return;
  const int c = i & 31;
  const int bv = i >> 5;
  const int b = bv / VV;
  const float mean = stats[(b * 32 + c) * 2 + 0];
  const float sd = stats[(b * 32 + c) * 2 + 1];
  const float g = h[(size_t)bv * 64 + c];
  const float be = h[(size_t)bv * 64 + 32 + c];
  t[i] = (1.0f + g) * ((x[i] - mean) / sd) + be;
}

// ---------------------------------------------------------------------------
// Host-side orchestration
// ---------------------------------------------------------------------------
extern "C" void kernel_launch(void* const* d_in, const int* in_sizes, int n_in,
                              void* d_out, int out_size, void* d_ws, size_t ws_size,
                              hipStream_t stream) {
  const float* src       = (const float*)d_in[0];   // (B,V,3)
  const float* tgt       = (const float*)d_in[1];   // (B,V,10)
  const float* conv0_d   = (const float*)d_in[2];   // (3,64)
  const float* conv1_w   = (const float*)d_in[3];   // (16,160)
  const float* conv1_b   = (const float*)d_in[4];   // (160)
  const float* conv1_d   = (const float*)d_in[5];   // (3,128)
  const float* adain_w   = (const float*)d_in[6];   // (10,64)
  const float* adain_b   = (const float*)d_in[7];   // (64)
  const float* dc1_w     = (const float*)d_in[8];   // (32,80)
  const float* dc1_b     = (const float*)d_in[9];   // (80)
  const float* dc1_d     = (const float*)d_in[10];  // (3,64)
  const float* dc2_w     = (const float*)d_in[11];  // (16,15)
  const float* dc2_b     = (const float*)d_in[12];  // (15)
  const float* dc2_d     = (const float*)d_in[13];  // (3,12)
  float* out = (float*)d_out;                       // (B,V,3)

  float* w = (float*)d_ws;
  size_t o = 0;
  float* dirs0n  = w + o; o += 3 * 64;
  float* dirs1n  = w + o; o += 3 * 128;
  float* dirsd1n = w + o; o += 3 * 64;
  float* dirsd2n = w + o; o += 3 * 12;
  o = (o + 15) & ~(size_t)15;
  int*   idx     = (int*)(w + o); o += (size_t)BV * NB;
  float* f1      = w + o; o += (size_t)BV * 16;
  float* feat1   = w + o; o += (size_t)BV * 160;
  float* f2      = w + o; o += (size_t)BV * 32;
  float* h       = w + o; o += (size_t)BV * 64;
  float* t       = w + o; o += (size_t)BV * 32;
  float* stats   = w + o; o += (size_t)BB * 32 * 2;
  float* featd1  = w + o; o += (size_t)BV * 80;
  float* c1      = w + o; o += (size_t)BV * 16;
  float* featd2  = w + o; o += (size_t)BV * 15;
  (void)ws_size; (void)n_in; (void)in_sizes; (void)out_size;

  // 0) normalize direction banks (axis-0)
  normalize_cols_kernel<<<1, 128, 0, stream>>>(conv0_d, dirs0n, 64);
  normalize_cols_kernel<<<1, 128, 0, stream>>>(conv1_d, dirs1n, 128);
  normalize_cols_kernel<<<1, 128, 0, stream>>>(dc1_d, dirsd1n, 64);
  normalize_cols_kernel<<<1, 128, 0, stream>>>(dc2_d, dirsd2n, 12);

  // 1) kNN (WMMA Gram tiles + LDS band + wave32 argmin)
  {
    dim3 grid(BB * (VV / 16));
    size_t shmem = (size_t)(16 * VV + 16) * sizeof(float);
    knn_kernel<<<grid, 256, shmem, stream>>>(src, idx);
  }

  // 2) f1 = relu(conv_surface)
  conv_surface_kernel<<<BV / 16, 256, 0, stream>>>(src, idx, dirs0n, f1);

  // 3) feat1 = f1 @ conv1_w + conv1_b   (K=16, N=160)
  gemm_bias_kernel<16, 160><<<(BV / 16) * 10, 32, 0, stream>>>(f1, conv1_w, conv1_b, feat1);

  // 4) f2 = relu(conv_layer(feat1, outc=32))
  {
    dim3 blk(32, 8);
    conv_layer_kernel<<<BV / 8, blk, 8 * NB * 3 * sizeof(float), stream>>>(
        src, idx, dirs1n, feat1, f2, 1);
  }

  // 5) h = target_feature @ adain_w + adain_b  (K=10, N=64)
  gemm_bias_kernel<10, 64><<<(BV / 16) * 4, 32, 0, stream>>>(tgt, adain_w, adain_b, h);

  // 6) AdaIN
  adain_stats_kernel<<<BB * 32, 256, 0, stream>>>(f2, stats);
  adain_apply_kernel<<<(BV * 32 + 255) / 256, 256, 0, stream>>>(f2, h, stats, t);

  // 7) featd1 = t @ dc1_w + dc1_b   (K=32, N=80)
  gemm_bias_kernel<32, 80><<<(BV / 16) * 5, 32, 0, stream>>>(t, dc1_w, dc1_b, featd1);

  // 8) c1 = relu(conv_layer(featd1, outc=16))
  {
    dim3 blk(16, 16);
    conv_layer_kernel<<<BV / 16, blk, 16 * NB * 3 * sizeof(float), stream>>>(
        src, idx, dirsd1n, featd1, c1, 1);
  }

  // 9) featd2 = c1 @ dc2_w + dc2_b   (K=16, N=15)
  gemm_bias_kernel<16, 15><<<(BV / 16) * 1, 32, 0, stream>>>(c1, dc2_w, dc2_b, featd2);

  // 10) out = sigmoid(conv_layer(featd2, outc=3))
  {
    dim3 blk(3, 64);
    conv_layer_kernel<<<BV / 64, blk, 64 * NB * 3 * sizeof(float), stream>>>(
        src, idx, dirsd2n, featd2, out, 2);
  }
}